// matchEngine_35768487641323
// MI455X (gfx1250) — compile-verified
//
#include <hip/hip_runtime.h>
#include <hip/hip_bf16.h>

// ---------------------------------------------------------------------------
// AES S-box (permutation of 0..255), baked into the code object.
// ---------------------------------------------------------------------------
__device__ __constant__ unsigned char d_sbox[256] = {
  99,124,119,123,242,107,111,197, 48,  1,103, 43,254,215,171,118,
 202,130,201,125,250, 89, 71,240,173,212,162,175,156,164,114,192,
 183,253,147, 38, 54, 63,247,204, 52,165,229,241,113,216, 49, 21,
   4,199, 35,195, 24,150,  5,154,  7, 18,128,226,235, 39,178,117,
   9,131, 44, 26, 27,110, 90,160, 82, 59,214,179, 41,227, 47,132,
  83,209,  0,237, 32,252,177, 91,106,203,190, 57, 74, 76, 88,207,
 208,239,170,251, 67, 77, 51,133, 69,249,  2,127, 80, 60,159,168,
  81,163, 64,143,146,157, 56,245,188,182,218, 33, 16,255,243,210,
 205, 12, 19,236, 95,151, 68, 23,196,167,126, 61,100, 93, 25,115,
  96,129, 79,220, 34, 42,144,136, 70,238,184, 20,222, 94, 11,219,
 224, 50, 58, 10, 73,  6, 36, 92,194,211,172, 98,145,149,228,121,
 231,200, 55,109,141,213, 78,169,108, 86,244,234,101,122,174,  8,
 186,120, 37, 46, 28,166,180,198,232,221,116, 31, 75,189,139,138,
 112, 62,181,102, 72,  3,246, 14, 97, 53, 87,185,134,193, 29,158,
 225,248,152, 17,105,217,142,148,155, 30,135,233,206, 85, 40,223,
 140,161,137, 13,191,230, 66,104, 65,153, 45, 15,176, 84,187, 22
};

#define BLOCK     256     // 8 wave32 waves
#define TILE_ROWS 4       // 4 rows x 256 f32 = 4 KB tile = 256 threads x b128
#define CFIX      256     // row length (predictions is (N,256))
#define NSEG      64      // segments for the parallel cumsum

// ---------------------------------------------------------------------------
// CDNA5 async global->LDS copy (ASYNCcnt-tracked, no VGPR round trip).
// Generic pointers to __shared__ carry the LDS byte offset in their low 32
// bits, which is exactly the VDST operand of global_load_async_to_lds_*.
// ---------------------------------------------------------------------------
__device__ __forceinline__ unsigned lds_off_of(const void* p) {
  return (unsigned)(unsigned long long)p;
}

__device__ __forceinline__ void async_copy_b128(unsigned lds_off, const float* g) {
  asm volatile("global_load_async_to_lds_b128 %0, %1, off"
               :: "v"(lds_off), "v"((unsigned long long)g)
               : "memory");
}

__device__ __forceinline__ void wait_async_le1() {
  asm volatile("s_wait_asynccnt 0x1" ::: "memory");
}
__device__ __forceinline__ void wait_async_0() {
  asm volatile("s_wait_asynccnt 0x0" ::: "memory");
}

// ---------------------------------------------------------------------------
// K1: per-chunk log-likelihood sums. One block processes one chunk at a time
// (persistent grid). Rows streamed into LDS with double-buffered async b128
// copies; each thread g gathers row[sbox[m^g]] from LDS and accumulates log.
// ---------------------------------------------------------------------------
__global__ void __launch_bounds__(BLOCK)
chunk_sums_kernel(const float* __restrict__ pred,
                  const int*   __restrict__ meta,
                  const int*   __restrict__ p_grange,
                  const int*   __restrict__ p_step,
                  float*       __restrict__ sums,
                  int N) {
  __shared__ float buf[2][TILE_ROWS * CFIX];   // 8 KB double buffer
  __shared__ int   sbox_lds[256];              // 1 KB

  const int tid    = threadIdx.x;
  const int grange = *p_grange;
  const int step   = *p_step;
  const int num_chunks = N / step;

  sbox_lds[tid] = (int)d_sbox[tid & 255];
  __syncthreads();

  const int num_tiles = (step + TILE_ROWS - 1) / TILE_ROWS;
  // each thread owns 16 bytes (4 f32) of the 4 KB tile
  const int e4 = tid * 4;            // element offset within tile
  const int tr = e4 >> 8;            // row within tile (0..3)
  const int tc = e4 & 255;           // column within row

  for (int k = blockIdx.x; k < num_chunks; k += gridDim.x) {
    const int chunk_base = k * step;

    // prologue: issue tile 0 into buf[0]  (1 async instr per wave slice)
    {
      int row = chunk_base + tr;
      row = row < N ? row : N - 1;             // clamp tail overshoot
      async_copy_b128(lds_off_of(&buf[0][e4]),
                      pred + (unsigned long long)row * CFIX + tc);
    }

    float acc = 0.0f;
    for (int t = 0; t < num_tiles; ++t) {
      const int cur = t & 1;
      if (t + 1 < num_tiles) {
        int row = chunk_base + (t + 1) * TILE_ROWS + tr;
        row = row < N ? row : N - 1;
        async_copy_b128(lds_off_of(&buf[cur ^ 1][e4]),
                        pred + (unsigned long long)row * CFIX + tc);
        wait_async_le1();                      // tile t landed, t+1 in flight
      } else {
        wait_async_0();                        // last tile landed
      }
      __syncthreads();                         // all waves' slices visible

      const int rmax = min(TILE_ROWS, step - t * TILE_ROWS);
      for (int r = 0; r < rmax; ++r) {
        const int m = meta[chunk_base + t * TILE_ROWS + r];  // block-uniform
        if (tid < grange) {
          const int   idx = sbox_lds[(m ^ tid) & 255];
          const float v   = buf[cur][r * CFIX + idx];
          acc += (v != 0.0f) ? logf(v) : 0.0f;
        }
      }
      __syncthreads();                         // safe to overwrite buf[cur]
    }

    if (tid < grange)
      sums[(unsigned long long)k * grange + tid] = acc;
  }
}

// ---------------------------------------------------------------------------
// K2: per-segment column sums (64 segments over the chunk axis).
// ---------------------------------------------------------------------------
__global__ void __launch_bounds__(BLOCK)
segsum_kernel(const float* __restrict__ sums, float* __restrict__ seg,
              const int* __restrict__ p_grange, const int* __restrict__ p_step,
              int N) {
  const int grange = *p_grange;
  const int step   = *p_step;
  const int num_chunks = N / step;
  const int cps = (num_chunks + NSEG - 1) / NSEG;
  const int b = blockIdx.x, g = threadIdx.x;
  const int k0 = b * cps;
  const int k1 = min(k0 + cps, num_chunks);
  float s = 0.0f;
  if (g < grange)
    for (int k = k0; k < k1; ++k)
      s += sums[(unsigned long long)k * grange + g];
  seg[b * 256 + g] = s;
}

// ---------------------------------------------------------------------------
// K3: exclusive prefix over the 64 segment totals (per guess column).
// ---------------------------------------------------------------------------
__global__ void __launch_bounds__(BLOCK)
segscan_kernel(float* __restrict__ seg) {
  const int g = threadIdx.x;
  float run = 0.0f;
  for (int b = 0; b < NSEG; ++b) {
    const float t = seg[b * 256 + g];
    seg[b * 256 + g] = run;
    run += t;
  }
}

// ---------------------------------------------------------------------------
// K4: finish cumsum inside each segment, rank the correct key per chunk via
// wave32 ballot+popcount, emit (ranks, x_rank) as floats.
// ---------------------------------------------------------------------------
__global__ void __launch_bounds__(BLOCK)
rank_kernel(const float* __restrict__ sums, const float* __restrict__ seg,
            const int* __restrict__ p_grange, const int* __restrict__ p_ck,
            const int* __restrict__ p_step, float* __restrict__ out, int N) {
  const int grange = *p_grange;
  const int ck     = *p_ck;
  const int step   = *p_step;
  const int num_chunks = N / step;
  const int cps = (num_chunks + NSEG - 1) / NSEG;
  const int b = blockIdx.x, g = threadIdx.x;
  const int k0 = b * cps;
  const int k1 = min(k0 + cps, num_chunks);

  __shared__ float key_score;
  __shared__ int   cnt[BLOCK / 32];

  float run = (g < grange) ? seg[b * 256 + g] : 0.0f;
  for (int k = k0; k < k1; ++k) {
    if (g < grange) run += sums[(unsigned long long)k * grange + g];
    if (g == ck) key_score = run;
    __syncthreads();
    const bool gt = (g < grange) && (run > key_score);
    const unsigned long long ball = __ballot(gt);   // wave32: low 32 bits
    if ((g & 31) == 0) cnt[g >> 5] = __popcll(ball);
    __syncthreads();
    if (g == 0) {
      int r = 0;
      for (int w = 0; w < BLOCK / 32; ++w) r += cnt[w];
      out[k] = (float)r;                            // ranks
      out[num_chunks + k] = (float)((k + 1) * step);// x_rank
    }
    __syncthreads();
  }
}

// ---------------------------------------------------------------------------
// Host launcher. Inputs: predictions(f32 N*256), metadata(i32 N),
// guess_range(i32[1]), correct_key(i32[1]), step(i32[1]). Scalars live on
// device, so all chunk math is device-side; grids are fixed (capture-safe).
// Workspace layout: [0,64KB) segment buffer, [64KB,...) chunk_sums.
// ---------------------------------------------------------------------------
extern "C" void kernel_launch(void* const* d_in, const int* in_sizes, int n_in,
                              void* d_out, int out_size, void* d_ws, size_t ws_size,
                              hipStream_t stream) {
  const float* pred   = (const float*)d_in[0];
  const int*   meta   = (const int*)  d_in[1];
  const int*   grange = (const int*)  d_in[2];
  const int*   ckey   = (const int*)  d_in[3];
  const int*   step   = (const int*)  d_in[4];
  const int N = in_sizes[1];                       // number of traces

  float* seg  = (float*)d_ws;                                   // 64*256 f32
  float* sums = (float*)((char*)d_ws + NSEG * 256 * sizeof(float));
  float* out  = (float*)d_out;

  chunk_sums_kernel<<<1024, BLOCK, 0, stream>>>(pred, meta, grange, step, sums, N);
  segsum_kernel    <<<NSEG, BLOCK, 0, stream>>>(sums, seg, grange, step, N);
  segscan_kernel   <<<1,    BLOCK, 0, stream>>>(seg);
  rank_kernel      <<<NSEG, BLOCK, 0, stream>>>(sums, seg, grange, ckey, step, out, N);
}